// WeightedSmoothLoss_21466246545456
// MI455X (gfx1250) — compile-verified
//
#include <hip/hip_runtime.h>

// Problem constants (match reference)
#define B_ROWS 524288
#define C_CLS  128
#define SMOOTH 0.1f

// Launch config: one wave per row, grid-stride over rows.
#define NBLK   2048
#define THREADS 256
#define WAVES_PER_BLOCK (THREADS / 32)
#define TOTAL_WAVES (NBLK * WAVES_PER_BLOCK)

typedef __attribute__((ext_vector_type(2))) float v2f;
typedef __attribute__((ext_vector_type(4))) float v4f;
typedef __attribute__((ext_vector_type(8))) float v8f;

// ---------------------------------------------------------------------------
// Stage 1: streaming loss kernel. One wave32 handles one row of 128 logits
// (lane loads float4 => 512B/row, fully coalesced, non-temporal since the
// 256MB stream exceeds the 192MB L2 and is touched exactly once).
// ---------------------------------------------------------------------------
__global__ __launch_bounds__(THREADS) void ce_loss_main(
    const float* __restrict__ logits,
    const int*   __restrict__ targets,
    const float* __restrict__ weight,
    const float* __restrict__ penalty,   // [C_CLS * C_CLS]
    float*       __restrict__ partials)  // [NBLK]
{
    const int lane  = threadIdx.x & 31;
    const int wave  = threadIdx.x >> 5;
    const int gwave = blockIdx.x * WAVES_PER_BLOCK + wave;

    float acc = 0.0f;

    for (int row = gwave; row < B_ROWS; row += TOTAL_WAVES) {
        const v4f* rp = (const v4f*)(logits + (size_t)row * C_CLS) + lane;
        v4f x = __builtin_nontemporal_load(rp);   // global_load_b128, TH=NT

        // ---- local max / first-argmax over the 4 elements -----------------
        float m  = x.x; int mi = lane * 4;
        if (x.y > m) { m = x.y; mi = lane * 4 + 1; }
        if (x.z > m) { m = x.z; mi = lane * 4 + 2; }
        if (x.w > m) { m = x.w; mi = lane * 4 + 3; }

        // ---- wave32 butterfly: lexicographic (max value, min index) -------
#pragma unroll
        for (int off = 16; off > 0; off >>= 1) {
            float om = __shfl_xor(m,  off, 32);
            int   oi = __shfl_xor(mi, off, 32);
            if (om > m || (om == m && oi < mi)) { m = om; mi = oi; }
        }

        // ---- sum(x) and sum(exp(x - max)) ---------------------------------
        float sx = x.x + x.y + x.z + x.w;
        float se = __expf(x.x - m) + __expf(x.y - m) +
                   __expf(x.z - m) + __expf(x.w - m);
#pragma unroll
        for (int off = 16; off > 0; off >>= 1) {
            sx += __shfl_xor(sx, off, 32);
            se += __shfl_xor(se, off, 32);
        }
        float lse = __logf(se);   // log_p[j] = x[j] - m - lse

        // ---- gather x[target] (target is wave-uniform) --------------------
        int t        = targets[row];
        int src_lane = t >> 2;
        int comp     = t & 3;
        float cand = (comp == 0) ? x.x : (comp == 1) ? x.y
                   : (comp == 2) ? x.z : x.w;
        float xt = __shfl(cand, src_lane, 32);

        float logp_t = xt - m - lse;
        float nll    = -logp_t * weight[t];
        // sum_j log_p[j] = sx - C*(m + lse)  => smooth = -mean(log_p)
        float smooth = -(sx - (float)C_CLS * (m + lse)) * (1.0f / (float)C_CLS);
        float loss   = (1.0f - SMOOTH) * nll + SMOOTH * smooth;
        float mask   = penalty[t * C_CLS + mi];   // mi == argmax (pred)
        acc += loss * mask;   // identical on all lanes; lane 0's copy is used
    }

    // ---- per-block deterministic reduction --------------------------------
    __shared__ float wpart[WAVES_PER_BLOCK];
    if (lane == 0) wpart[wave] = acc;
    __syncthreads();
    if (threadIdx.x == 0) {
        float s = 0.0f;
#pragma unroll
        for (int i = 0; i < WAVES_PER_BLOCK; ++i) s += wpart[i];
        partials[blockIdx.x] = s;
    }
}

// ---------------------------------------------------------------------------
// Stage 2: single-wave final reduction of the 2048 block partials using
// V_WMMA_F32_16X16X4_F32 with B = all-ones: D = A x 1 + C accumulates exact
// f32 row sums (64 values per WMMA, 32 iterations).
//
// A layout (32-bit 16x4, ISA 7.12.2): lane L holds M = L%16; VGPR0 = K=(L<16?0:2),
// VGPR1 = K+1. We map A[m][k] = partials[base + m*4 + k], i.e. a strided
// 2-float load per lane. After accumulation, D column 0 lives in lanes 0
// (M=0..7, VGPRs 0..7) and 16 (M=8..15), so total = sum8(lane0) + sum8(lane16).
// ---------------------------------------------------------------------------
__global__ __launch_bounds__(32) void ce_loss_reduce(
    const float* __restrict__ partials, float* __restrict__ out)
{
    const int lane = threadIdx.x;   // full wave32, EXEC all ones (WMMA req)

    v2f b; b.x = 1.0f; b.y = 1.0f;  // B = ones(4x16), layout-independent
    v8f c = {};                     // f32 accumulator (C/D matrix)

#pragma unroll 4
    for (int it = 0; it < NBLK / 64; ++it) {
        int base = it * 64;
        int idx  = base + (lane & 15) * 4 + ((lane >> 4) << 1);
        v2f a;
        a.x = partials[idx];
        a.y = partials[idx + 1];
        // (neg_a, A, neg_b, B, c_mod, C, reuse_a, reuse_b)
        c = __builtin_amdgcn_wmma_f32_16x16x4_f32(
                false, a, false, b, (short)0, c, false, false);
    }

    float s = c[0] + c[1] + c[2] + c[3] + c[4] + c[5] + c[6] + c[7];
    float tot = __shfl(s, 0, 32) + __shfl(s, 16, 32);
    if (lane == 0) out[0] = tot * (1.0f / (float)B_ROWS);
}

// ---------------------------------------------------------------------------
extern "C" void kernel_launch(void* const* d_in, const int* in_sizes, int n_in,
                              void* d_out, int out_size, void* d_ws, size_t ws_size,
                              hipStream_t stream) {
    const float* logits  = (const float*)d_in[0];
    const int*   targets = (const int*)  d_in[1];
    const float* weight  = (const float*)d_in[2];
    const float* penalty = (const float*)d_in[3];
    float*       out     = (float*)d_out;
    float*       partials = (float*)d_ws;   // NBLK * 4 = 8 KB scratch

    ce_loss_main<<<NBLK, THREADS, 0, stream>>>(logits, targets, weight, penalty, partials);
    ce_loss_reduce<<<1, 32, 0, stream>>>(partials, out);
}